// MixJKLayer_14697378087201
// MI455X (gfx1250) — compile-verified
//
#include <hip/hip_runtime.h>

#define D 128
#define XS_STRIDE 132   // 132 % 64 = 4  -> A-operand LDS reads conflict-free
#define W_STRIDE  136   // 2*136 % 64 = 16 -> B-operand lane halves hit disjoint banks
#define RPB 80          // rows per block: 100000 = 1250 * 80, no tail
#define RT  5           // row tiles (16 rows each) per wave
#define BETA 0.5f
#define CCONST 1.0f

typedef __attribute__((ext_vector_type(2))) float v2f;
typedef __attribute__((ext_vector_type(8))) float v8f;

// gfx1250 async global->LDS copy (ASYNCcnt-tracked, bypasses VGPRs).
// GVS mode: mem = SGPR_base + VGPR_offset; vdst = LDS byte address.
__device__ __forceinline__ void async_copy_b128(unsigned lds_addr,
                                                const void* sbase,
                                                unsigned goff) {
  asm volatile("global_load_async_to_lds_b128 %0, %1, %2"
               :: "v"(lds_addr), "v"(goff), "s"(sbase)
               : "memory");
}

__device__ __forceinline__ void async_wait_all() {
  asm volatile("s_wait_asynccnt 0x0" ::: "memory");
}

// -------------------- Stage 1: h = x @ W via V_WMMA_F32_16X16X4_F32 --------------------
// Block = 256 threads = 8 waves, 80 rows x 128 cols of h per block.
// Wave w owns the 16-column tile [16w,16w+16) and accumulates 5 row-tiles, so each
// B-operand LDS load feeds 5 WMMAs. fp32 end-to-end (matches the f32 reference).
__global__ __launch_bounds__(256)
void gcn_gemm_wmma_f32(const float* __restrict__ x,
                       const float* __restrict__ W,
                       float* __restrict__ h) {
  __shared__ float Wlds[D * W_STRIDE];        // ~69.6 KB
  __shared__ float xs[RPB * XS_STRIDE];       // ~42.2 KB

  const int tid = threadIdx.x;
  const long rowBase = (long)blockIdx.x * RPB;

  // Async-stage W (128x128 f32): 4096 b128 transfers, LDS rows padded to W_STRIDE.
  for (int idx = tid; idx < D * (D / 4); idx += 256) {
    const int r = idx >> 5;                   // row 0..127
    const int q = idx & 31;                   // float4 slot
    async_copy_b128((unsigned)(uintptr_t)&Wlds[r * W_STRIDE + q * 4],
                    W, (unsigned)((r * D + q * 4) * sizeof(float)));
  }
  // Async-stage the 80-row x slab: 2560 b128 transfers.
  const float* xslab = x + rowBase * D;
  for (int idx = tid; idx < RPB * (D / 4); idx += 256) {
    const int r = idx >> 5;
    const int q = idx & 31;
    async_copy_b128((unsigned)(uintptr_t)&xs[r * XS_STRIDE + q * 4],
                    xslab, (unsigned)((r * D + q * 4) * sizeof(float)));
  }
  async_wait_all();
  __syncthreads();

  const int lane = tid & 31;
  const int wave = tid >> 5;        // 0..7 -> 16-column tile
  const int m    = lane & 15;
  const int hi   = lane >> 4;       // lane half: K{0,1} vs K{2,3} / M vs M+8
  const int koff = hi * 2;
  const int col  = wave * 16 + m;

  // ISA f32 layouts: A 16x4 -> VGPR0 = K=(0|2), VGPR1 = K=(1|3) by lane half.
  //                  B 4x16 -> same K split, row striped across lanes.
  v8f acc[RT] = {};
  for (int k0 = 0; k0 < D; k0 += 4) {
    v2f bm;
    bm.x = Wlds[(k0 + koff) * W_STRIDE + col];
    bm.y = Wlds[(k0 + koff + 1) * W_STRIDE + col];
    #pragma unroll
    for (int t = 0; t < RT; ++t) {
      v2f a;
      a.x = xs[(t * 16 + m) * XS_STRIDE + k0 + koff];
      a.y = xs[(t * 16 + m) * XS_STRIDE + k0 + koff + 1];
      acc[t] = __builtin_amdgcn_wmma_f32_16x16x4_f32(
          /*neg_a=*/false, a, /*neg_b=*/false, bm,
          /*c_mod=*/(short)0, acc[t], /*reuse_a=*/false, /*reuse_b=*/false);
    }
  }

  // D layout: VGPR r -> row (r + 8*hi). Exact-division grid: no guards, EXEC stays full.
  float* hbase = h + rowBase * D + col;
  #pragma unroll
  for (int t = 0; t < RT; ++t) {
    #pragma unroll
    for (int r = 0; r < 8; ++r) {
      hbase[(long)(t * 16 + r + hi * 8) * D] = acc[t][r];
    }
  }
}

// -------------------- Stage 0: zero the accumulator (d_out) --------------------
__global__ void gcn_zero_f32(float* __restrict__ p, long n) {
  long i = (long)blockIdx.x * blockDim.x + threadIdx.x;
  const long stride = (long)gridDim.x * blockDim.x;
  for (; i < n; i += stride) p[i] = 0.0f;
}

// -------------------- Stage 2: edge gather * w -> atomic scatter-add --------------------
// One wave per edge; lane L owns float4 slice [4L,4L+4) of the 128-wide row.
// h (51 MB) and agg (51 MB) both fit in the 192 MB L2: gathers + 205M f32 atomics
// resolve in L2. This stage is the roofline limiter.
__global__ __launch_bounds__(256)
void gcn_edge_scatter(const float* __restrict__ h,
                      const long long* __restrict__ src,
                      const long long* __restrict__ dst,
                      const float* __restrict__ ew,
                      float* __restrict__ agg,
                      int n_edges) {
  const int edge = blockIdx.x * (blockDim.x >> 5) + (threadIdx.x >> 5);
  if (edge >= n_edges) return;
  const int lane = threadIdx.x & 31;

  const long s = (long)src[edge];
  const long d = (long)dst[edge];
  const float w = ew[edge];

  const float4* hrow = (const float4*)(h + s * D);
  __builtin_prefetch(hrow, 0, 1);            // global_prefetch_b8
  const float4 v = hrow[lane];

  float* a = agg + d * D + lane * 4;
  atomicAdd(a + 0, v.x * w);
  atomicAdd(a + 1, v.y * w);
  atomicAdd(a + 2, v.z * w);
  atomicAdd(a + 3, v.w * w);
}

// -------------------- Stage 3: z = agg + b; y = beta*z + (1-beta)*relu(z) --------------------
__global__ void gcn_epilogue(float* __restrict__ z, const float* __restrict__ b, long n4) {
  long i = (long)blockIdx.x * blockDim.x + threadIdx.x;
  const long stride = (long)gridDim.x * blockDim.x;
  const float4* bv4 = (const float4*)b;      // 32 float4 = 128 cols
  for (; i < n4; i += stride) {
    float4 zv = ((float4*)z)[i];
    const float4 bv = bv4[i & 31];
    float4 y;
    float t;
    t = zv.x + bv.x; y.x = BETA * t + (CCONST - BETA) * fmaxf(t, 0.0f);
    t = zv.y + bv.y; y.y = BETA * t + (CCONST - BETA) * fmaxf(t, 0.0f);
    t = zv.z + bv.z; y.z = BETA * t + (CCONST - BETA) * fmaxf(t, 0.0f);
    t = zv.w + bv.w; y.w = BETA * t + (CCONST - BETA) * fmaxf(t, 0.0f);
    ((float4*)z)[i] = y;
  }
}

extern "C" void kernel_launch(void* const* d_in, const int* in_sizes, int n_in,
                              void* d_out, int out_size, void* d_ws, size_t ws_size,
                              hipStream_t stream) {
  const float*      x  = (const float*)d_in[0];      // [N, 128] f32
  const long long*  ei = (const long long*)d_in[1];  // [2, E] i64 (row0=src, row1=dst)
  const float*      ew = (const float*)d_in[2];      // [E] f32
  const float*      W  = (const float*)d_in[3];      // [128, 128] f32
  const float*      b  = (const float*)d_in[4];      // [128] f32
  float* out = (float*)d_out;                        // [N, 128] f32 (accumulator, then y)
  float* h   = (float*)d_ws;                         // [N, 128] f32 scratch (51.2 MB)

  const int N = in_sizes[0] / D;        // 100000 (multiple of RPB=80)
  const int E = in_sizes[2];            // 1600000

  // Stage 0: zero the scatter accumulator (d_out).
  gcn_zero_f32<<<2048, 256, 0, stream>>>(out, (long)N * D);

  // Stage 1: h = x @ W (fp32 WMMA), 80 rows per block, 1250 blocks exact.
  gcn_gemm_wmma_f32<<<N / RPB, 256, 0, stream>>>(x, W, h);

  // Stage 2: per-edge gather/scale/atomic-scatter, 8 edges per 256-thread block.
  gcn_edge_scatter<<<(E + 7) / 8, 256, 0, stream>>>(h, ei, ei + E, ew, out, E);

  // Stage 3: bias + activation mix, in place on d_out.
  gcn_epilogue<<<2048, 256, 0, stream>>>(out, b, (long)N * D / 4);
}